// HolographicConv2d_47330539602434
// MI455X (gfx1250) — compile-verified
//
#include <hip/hip_runtime.h>
#include <hip/hip_bf16.h>

// ---------------------------------------------------------------------------
// HolographicConv2d on gfx1250:
//   y[b,o] = IFFT2( sum_i FFT2(x[b,i]) * W[o,i] )
// Stage 1: FFT2(x)  -> x_freq (workspace), Stockham radix-2 in LDS
// Stage 2: per-bin complex GEMM [16x128]x[128x128] via V_WMMA_F32_16X16X4_F32.
//          W (536 MB, used exactly once -> ~23 us floor @ 23.3 TB/s) streamed
//          into LDS by the Tensor Data Mover (TENSOR_LOAD_TO_LDS, TENSORcnt),
//          double-buffered; TDM pad_interval/amount reproduces the padded
//          bank-conflict-free LDS layout in hardware.
// Stage 3: IFFT2 in place on d_out (scale 1/4096)
// ---------------------------------------------------------------------------

#define B_SZ    16
#define C_IN    128
#define C_OUT   128
#define GRID_N  64
#define NBINS   (GRID_N * GRID_N)      // 4096
#define ROWP    65                     // padded LDS row pitch (complex)
#define WPITCH  17                     // bank-conflict pad for 16-bin tiles

typedef __attribute__((ext_vector_type(2))) float v2f;
typedef __attribute__((ext_vector_type(8))) float v8f;

typedef int v2i __attribute__((ext_vector_type(2)));
typedef __attribute__((address_space(1))) v2i* gptr_b64;
typedef __attribute__((address_space(3))) v2i* lptr_b64;

typedef unsigned int v4u  __attribute__((ext_vector_type(4)));
typedef int          v8i_t __attribute__((ext_vector_type(8)));
typedef int          v4i_t __attribute__((ext_vector_type(4)));

#if __has_builtin(__builtin_amdgcn_tensor_load_to_lds)
#define USE_TDM 1
#else
#define USE_TDM 0
#endif

// --- CDNA5 async global<->LDS helpers (ASYNCcnt path) ----------------------
__device__ __forceinline__ void async_g2l_b64(const void* g, void* l) {
#if __has_builtin(__builtin_amdgcn_global_load_async_to_lds_b64)
    __builtin_amdgcn_global_load_async_to_lds_b64(
        (gptr_b64)(v2i*)(void*)g, (lptr_b64)(v2i*)l, 0, 0);
#else
    *(float2*)l = *(const float2*)g;
#endif
}

__device__ __forceinline__ void async_l2g_b64(void* g, const void* l) {
#if __has_builtin(__builtin_amdgcn_global_store_async_from_lds_b64)
    __builtin_amdgcn_global_store_async_from_lds_b64(
        (gptr_b64)(v2i*)g, (lptr_b64)(v2i*)(void*)l, 0, 0);
#else
    *(float2*)g = *(const float2*)l;
#endif
}

__device__ __forceinline__ void wait_async0() {
#if __has_builtin(__builtin_amdgcn_s_wait_asynccnt)
    __builtin_amdgcn_s_wait_asynccnt(0);
#else
    asm volatile("s_wait_asynccnt 0x0" ::: "memory");
#endif
}

// --- Tensor Data Mover: 3-D tile load, x = 16 complex bins (128B rows),
// y/z arbitrary count+stride (complex units). LDS padded by 1 complex per
// 16-complex row (pad_interval code 4 = 32 DWORDs, pad_amount code 1 = 2 DW)
// == the WPITCH=17 layout. Issued by one wave; tracked with TENSORcnt.
#if USE_TDM
__device__ __forceinline__ void tdm_issue(const float2* gbase, const void* ldsPtr,
                                          unsigned tileY, unsigned tileZ) {
    const unsigned long long ga = (unsigned long long)(uintptr_t)gbase;
    const unsigned ldsOff = (unsigned)(size_t)(lptr_b64)(v2i*)(void*)ldsPtr;
    const unsigned long long sy = (unsigned long long)C_IN * NBINS;  // 524288
    const unsigned long long sz = NBINS;                             // 4096
    // D# group 0: count=1 | lds_addr | global_addr[56:0] | type=2
    const v4u g0 = {
        1u,
        ldsOff,
        (unsigned)(ga & 0xffffffffull),
        (unsigned)((ga >> 32) & 0x01ffffffull) | (2u << 30)
    };
    // D# group 1: data_size=3(8B), pad_enable, pad_interval=4, pad_amount=1;
    // tensor_dim0=16, tensor_dim1=tileY; tile_dim0=16, tile_dim1=tileY,
    // tile_dim2=tileZ; dim0_stride=sy, dim1_stride=sz (48-bit each).
    const v8i_t g1 = {
        (int)((3u << 16) | (1u << 20) | (4u << 22) | (1u << 25)),
        (int)(16u << 16),                          // tensor_dim0 lo16 @ [63:48]
        (int)(tileY << 16),                        // tensor_dim1 lo16 @ [95:80]
        (int)(16u << 16),                          // tile_dim0 @ [127:112]
        (int)(tileY | (tileZ << 16)),              // tile_dim1 | tile_dim2
        (int)(sy & 0xffffffffull),                 // dim0_stride [31:0]
        (int)(((sy >> 32) & 0xffffull) | ((sz & 0xffffull) << 16)),
        (int)((sz >> 16) & 0xffffffffull)          // dim1_stride [47:16]
    };
    const v4i_t g2 = { (int)tileZ, 0, 0, 0 };      // tensor_dim2
    const v4i_t g3 = { 0, 0, 0, 0 };
    const v8i_t g4 = { 0, 0, 0, 0, 0, 0, 0, 0 };   // unused (zero-filled)
    __builtin_amdgcn_tensor_load_to_lds(g0, g1, g2, g3, g4, 0);
}
#endif

__device__ __forceinline__ float2 cmulf2(float2 a, float2 b) {
    return make_float2(a.x * b.x - a.y * b.y, a.x * b.y + a.y * b.x);
}

// One 64-point Stockham radix-2 pass over 64 "lines"; ls/es strides in
// complex units. 6 stages ping-pong A<->B, result ends in A. 256 threads.
__device__ void fft64_pass(float2* __restrict__ A, float2* __restrict__ B,
                           int ls, int es, float sign) {
    float2* s = A;
    float2* d = B;
    int l = 32;
    for (int st = 0; st < 6; ++st) {
        const int m = 1 << st;
        for (int idx = threadIdx.x; idx < 64 * 32; idx += 256) {
            const int line = idx >> 5;
            const int u    = idx & 31;
            const int j    = u >> st;
            const int kk   = u & (m - 1);
            float sn, cs;
            __sincosf(sign * 3.14159265358979323846f * (float)j / (float)l, &sn, &cs);
            const float2 w = make_float2(cs, sn);
            const int base = line * ls;
            const float2 c0 = s[base + (kk + j * m) * es];
            const float2 c1 = s[base + (kk + j * m + l * m) * es];
            const float2 sum = make_float2(c0.x + c1.x, c0.y + c1.y);
            const float2 dif = make_float2(c0.x - c1.x, c0.y - c1.y);
            d[base + (kk + 2 * j * m) * es]     = sum;
            d[base + (kk + 2 * j * m + m) * es] = cmulf2(dif, w);
        }
        __syncthreads();
        float2* t = s; s = d; d = t;
        l >>= 1;
    }
}

// ---------------------------------------------------------------------------
// Stage 1: forward FFT2 of real input x -> x_freq workspace
// ---------------------------------------------------------------------------
__global__ void __launch_bounds__(256)
fft2_fwd_kernel(const float* __restrict__ x, float2* __restrict__ xf) {
    extern __shared__ float2 smf[];
    float2* A = smf;
    float2* B = smf + GRID_N * ROWP;

    const int img = blockIdx.x;                       // b*C_IN + i
    const float* src = x + (size_t)img * NBINS;
    for (int t = threadIdx.x; t < NBINS; t += 256) {
        const int h = t >> 6, w = t & 63;
        A[h * ROWP + w] = make_float2(src[t], 0.0f);
    }
    __syncthreads();

    fft64_pass(A, B, ROWP, 1, -1.0f);   // rows (along w)
    fft64_pass(A, B, 1, ROWP, -1.0f);   // cols (along h)

    float2* dst = xf + (size_t)img * NBINS;
    for (int t = threadIdx.x; t < NBINS; t += 256) {
        const int h = t >> 6, w = t & 63;
        dst[t] = A[h * ROWP + w];
    }
}

// ---------------------------------------------------------------------------
// Stage 2: per-bin complex GEMM with fp32 WMMA + TDM double-buffered stream.
// grid = NBINS/16 blocks of 512 threads (16 waves); wave w owns one bin.
// ---------------------------------------------------------------------------
#define WCPLX  (4 * 128 * WPITCH)          // W chunk complex count (padded)
#define ACPLX  (4 * 16 * WPITCH)           // A chunk complex count (padded)
#define BUFC   (WCPLX + ACPLX)             // complex per double-buffer half

// Fallback (no TDM): per-thread async DMA for one chunk.
__device__ __forceinline__ void stage_chunk(const float2* __restrict__ xf,
                                            const float2* __restrict__ W,
                                            float2* __restrict__ Wl,
                                            float2* __restrict__ Al,
                                            int tid, int binbase, int i0) {
#pragma unroll
    for (int r = 0; r < 16; ++r) {
        const int f    = tid + r * 512;
        const int binI = f & 15;
        const int o    = (f >> 4) & 127;
        const int ic   = f >> 11;
        async_g2l_b64(&W[((size_t)(o * C_IN + i0 + ic)) * NBINS + binbase + binI],
                      &Wl[(ic * 128 + o) * WPITCH + binI]);
    }
#pragma unroll
    for (int r = 0; r < 2; ++r) {
        const int f    = tid + r * 512;
        const int binI = f & 15;
        const int b    = (f >> 4) & 15;
        const int ic   = f >> 8;
        async_g2l_b64(&xf[((size_t)(b * C_IN + i0 + ic)) * NBINS + binbase + binI],
                      &Al[(ic * 16 + b) * WPITCH + binI]);
    }
}

__global__ void __launch_bounds__(512)
freq_cgemm_kernel(const float2* __restrict__ xf, const float2* __restrict__ W,
                  float2* __restrict__ out) {
    extern __shared__ float2 smg[];
    float2* T = smg;                         // epilogue alias of buffer 0

    const int tid     = threadIdx.x;
    const int wave    = tid >> 5;
    const int lane    = tid & 31;
    const int laneN   = lane & 15;
    const int k0      = (lane >> 4) * 2;     // 16x4 A / 4x16 B lane->K mapping
    const int binbase = blockIdx.x * 16;
    const int q       = wave;                // bin within tile

    v8f accr[8], acci[8];
#pragma unroll
    for (int n = 0; n < 8; ++n) {
        accr[n] = (v8f){0.f, 0.f, 0.f, 0.f, 0.f, 0.f, 0.f, 0.f};
        acci[n] = (v8f){0.f, 0.f, 0.f, 0.f, 0.f, 0.f, 0.f, 0.f};
    }

    // Prologue: kick off chunk 0 into buffer 0.
#if USE_TDM
    if (wave == 0) {
        tdm_issue(&W[(size_t)0 * NBINS + binbase],  smg,         128, 4);
        tdm_issue(&xf[(size_t)0 * NBINS + binbase], smg + WCPLX, 16,  4);
    }
#else
    stage_chunk(xf, W, smg, smg + WCPLX, tid, binbase, 0);
#endif

    for (int c = 0; c < 32; ++c) {           // cin chunks of 4
        float2* Wl = smg + (size_t)(c & 1) * BUFC;
        float2* Al = Wl + WCPLX;

#if USE_TDM
        if (wave == 0) __builtin_amdgcn_s_wait_tensorcnt(0);
#else
        wait_async0();
#endif
        __syncthreads();                     // DMA visible to all waves; the
                                             // other buffer's consumers done
        if (c + 1 < 32) {
            float2* Wn = smg + (size_t)((c + 1) & 1) * BUFC;
            const int i1 = (c + 1) * 4;
#if USE_TDM
            if (wave == 0) {
                tdm_issue(&W[(size_t)i1 * NBINS + binbase],  Wn,         128, 4);
                tdm_issue(&xf[(size_t)i1 * NBINS + binbase], Wn + WCPLX, 16,  4);
            }
#else
            stage_chunk(xf, W, Wn, Wn + WCPLX, tid, binbase, i1);
#endif
        }

        // A fragments (16x4): lane m=laneN holds K = k0, k0+1
        const float2 a0 = Al[(k0 * 16 + laneN) * WPITCH + q];
        const float2 a1 = Al[((k0 + 1) * 16 + laneN) * WPITCH + q];
        const v2f Ar = {a0.x, a1.x};
        const v2f Ai = {a0.y, a1.y};

#pragma unroll
        for (int nt = 0; nt < 8; ++nt) {     // cout tiles of 16
            const int o = nt * 16 + laneN;
            const float2 w0 = Wl[(k0 * 128 + o) * WPITCH + q];
            const float2 w1 = Wl[((k0 + 1) * 128 + o) * WPITCH + q];
            const v2f Br  = {w0.x, w1.x};
            const v2f Bi  = {w0.y, w1.y};
            const v2f nBi = {-w0.y, -w1.y};  // fp32 WMMA has no A/B neg
            // Cr += Ar*Br - Ai*Bi ; Ci += Ar*Bi + Ai*Br
            accr[nt] = __builtin_amdgcn_wmma_f32_16x16x4_f32(
                false, Ar, false, Br, (short)0, accr[nt], false, false);
            accr[nt] = __builtin_amdgcn_wmma_f32_16x16x4_f32(
                false, Ai, false, nBi, (short)0, accr[nt], false, false);
            acci[nt] = __builtin_amdgcn_wmma_f32_16x16x4_f32(
                false, Ar, false, Bi, (short)0, acci[nt], false, false);
            acci[nt] = __builtin_amdgcn_wmma_f32_16x16x4_f32(
                false, Ai, false, Br, (short)0, acci[nt], false, false);
        }
    }

    // Epilogue: stage each cout tile through LDS (alias of buffer 0) so global
    // stores are 16-bin (128B) contiguous runs, issued as async LDS->global.
    // C/D layout: VGPR j -> M=j (lanes 0-15), M=j+8 (lanes 16-31); N = laneN.
    const int m0 = (lane >> 4) * 8;
    for (int nt = 0; nt < 8; ++nt) {
        __syncthreads();                     // prior tile's async reads waited on
#pragma unroll
        for (int j = 0; j < 8; ++j) {
            const int m = m0 + j;
            T[(m * 16 + laneN) * WPITCH + q] =
                make_float2(accr[nt][j], acci[nt][j]);
        }
        __syncthreads();
#pragma unroll
        for (int r = 0; r < 8; ++r) {
            const int f    = tid + r * 512;  // [0, 4096)
            const int binI = f & 15;
            const int oo   = (f >> 4) & 15;
            const int b    = f >> 8;
            async_l2g_b64(
                &out[((size_t)(b * C_OUT + nt * 16 + oo)) * NBINS + binbase + binI],
                &T[(b * 16 + oo) * WPITCH + binI]);
        }
        wait_async0();                       // T's data consumed before reuse
    }
}

// ---------------------------------------------------------------------------
// Stage 3: inverse FFT2, in place on d_out, scaled by 1/4096
// ---------------------------------------------------------------------------
__global__ void __launch_bounds__(256)
ifft2_kernel(float2* __restrict__ io) {
    extern __shared__ float2 smi[];
    float2* A = smi;
    float2* B = smi + GRID_N * ROWP;

    const int img = blockIdx.x;                      // b*C_OUT + o
    float2* p = io + (size_t)img * NBINS;
    for (int t = threadIdx.x; t < NBINS; t += 256) {
        const int h = t >> 6, w = t & 63;
        A[h * ROWP + w] = p[t];
    }
    __syncthreads();

    fft64_pass(A, B, ROWP, 1, +1.0f);
    fft64_pass(A, B, 1, ROWP, +1.0f);

    const float sc = 1.0f / (float)NBINS;
    for (int t = threadIdx.x; t < NBINS; t += 256) {
        const int h = t >> 6, w = t & 63;
        const float2 v = A[h * ROWP + w];
        p[t] = make_float2(v.x * sc, v.y * sc);
    }
}

// ---------------------------------------------------------------------------
extern "C" void kernel_launch(void* const* d_in, const int* in_sizes, int n_in,
                              void* d_out, int out_size, void* d_ws, size_t ws_size,
                              hipStream_t stream) {
    (void)in_sizes; (void)n_in; (void)out_size; (void)ws_size;

    const float*  x  = (const float*)d_in[0];                 // [16,128,64,64] f32
    const float2* W  = (const float2*)d_in[1];                // [128,128,64,64] c64
    float2*       y  = (float2*)d_out;                        // [16,128,64,64] c64
    float2*       xf = (float2*)d_ws;                         // 16*128*4096 c64 = 64 MiB

    const size_t fftLds  = (size_t)2 * GRID_N * ROWP * sizeof(float2);  // 66,560 B
    const size_t gemmLds = (size_t)2 * BUFC * sizeof(float2);           // 156,672 B

    fft2_fwd_kernel<<<B_SZ * C_IN, 256, fftLds, stream>>>(x, xf);
    freq_cgemm_kernel<<<NBINS / 16, 512, gemmLds, stream>>>(xf, W, y);
    ifft2_kernel<<<B_SZ * C_OUT, 256, fftLds, stream>>>(y);
}